// RbfLayer_24223615550020
// MI455X (gfx1250) — compile-verified
//
#include <hip/hip_runtime.h>
#include <math.h>

#define N_PTS   131072
#define N_CEN   128
#define KF      12          // 10 features padded to 12 (3 x K=4 WMMA slices)
#define PPB     64          // points per block
#define VSTR    132         // LDS stride (bank-friendly, 16B aligned)
#define BLOCK   128         // 4 wave32s

typedef float v2f __attribute__((ext_vector_type(2)));
typedef float v8f __attribute__((ext_vector_type(8)));

// ---------------------------------------------------------------------------
// Prep: per-center quadratic coefficients Q[128][12] + penalty init + bbox pen
// q(x) = Q0*x^2 + Q1*y^2 + Q2*z^2 + Q3*xy + Q4*xz + Q5*yz + Q6*x + Q7*y + Q8*z + Q9
// ---------------------------------------------------------------------------
__global__ __launch_bounds__(128)
void nerflet_prep(const float* __restrict__ centers,
                  const float* __restrict__ radii,
                  const float* __restrict__ rot,
                  float* __restrict__ Q,
                  float* __restrict__ pen)
{
    const int i = threadIdx.x;
    if (i == 0) *pen = 0.0f;
    __syncthreads();
    if (i >= N_CEN) return;

    const float c0 = centers[3*i+0], c1 = centers[3*i+1], c2 = centers[3*i+2];
    const float d0 = 1.0f / (fabsf(radii[3*i+0]) + 1e-8f);
    const float d1 = 1.0f / (fabsf(radii[3*i+1]) + 1e-8f);
    const float d2 = 1.0f / (fabsf(radii[3*i+2]) + 1e-8f);

    float sx, cx, sy, cy, sz, cz;
    __sincosf(rot[3*i+0], &sx, &cx);
    __sincosf(rot[3*i+1], &sy, &cy);
    __sincosf(rot[3*i+2], &sz, &cz);

    float R00 = cz*cy, R01 = cz*sy*sx - sz*cx, R02 = cz*sy*cx + sz*sx;
    float R10 = sz*cy, R11 = sz*sy*sx + cz*cx, R12 = sz*sy*cx - cz*sx;
    float R20 = -sy,   R21 = cy*sx,            R22 = cy*cx;

    // M[a][b] = sum_k R[a][k] * d[k] * R[b][k]  (symmetric)
    float m00 = R00*d0*R00 + R01*d1*R01 + R02*d2*R02;
    float m11 = R10*d0*R10 + R11*d1*R11 + R12*d2*R12;
    float m22 = R20*d0*R20 + R21*d1*R21 + R22*d2*R22;
    float m01 = R00*d0*R10 + R01*d1*R11 + R02*d2*R12;
    float m02 = R00*d0*R20 + R01*d1*R21 + R02*d2*R22;
    float m12 = R10*d0*R20 + R11*d1*R21 + R12*d2*R22;

    float g0 = -2.0f*(m00*c0 + m01*c1 + m02*c2);
    float g1 = -2.0f*(m01*c0 + m11*c1 + m12*c2);
    float g2 = -2.0f*(m02*c0 + m12*c1 + m22*c2);
    float cMc = m00*c0*c0 + m11*c1*c1 + m22*c2*c2
              + 2.0f*(m01*c0*c1 + m02*c0*c2 + m12*c1*c2);

    float* q = &Q[i*KF];
    q[0] = m00;      q[1] = m11;      q[2] = m22;
    q[3] = 2.0f*m01; q[4] = 2.0f*m02; q[5] = 2.0f*m12;
    q[6] = g0;       q[7] = g1;       q[8] = g2;
    q[9] = cMc;      q[10] = 0.0f;    q[11] = 0.0f;

    // bbox penalty: relu(c - 0.5) + relu(-0.5 - c), summed over all centers/dims
    float bb = fmaxf(c0-0.5f,0.f)+fmaxf(-0.5f-c0,0.f)
             + fmaxf(c1-0.5f,0.f)+fmaxf(-0.5f-c1,0.f)
             + fmaxf(c2-0.5f,0.f)+fmaxf(-0.5f-c2,0.f);
    atomicAdd(pen, bb);
}

// ---------------------------------------------------------------------------
// Main: WMMA rbf logits -> exp -> LDS tile -> per-point top16/threshold/blend
// ---------------------------------------------------------------------------
__global__ __launch_bounds__(BLOCK)
void nerflet_main(const float* __restrict__ xyz,
                  const float* __restrict__ Q,
                  const float* __restrict__ rgbs,
                  const float* __restrict__ dists,
                  float* __restrict__ out,
                  float* __restrict__ pen)
{
    __shared__ __align__(16) float sQ[N_CEN * KF];    // center coeffs
    __shared__ __align__(16) float sF[PPB * KF];      // point features (reused as reduce buf)
    __shared__ __align__(16) float sV[PPB * VSTR];    // rbf values [point][center]

    const int tid   = threadIdx.x;
    const int pbase = blockIdx.x * PPB;

    // stage Q
    for (int j = tid; j < N_CEN * KF; j += BLOCK) sQ[j] = Q[j];

    // stage point features
    if (tid < PPB) {
        const int p = pbase + tid;
        const float x = xyz[3*p+0], y = xyz[3*p+1], z = xyz[3*p+2];
        float* f = &sF[tid * KF];
        f[0] = x*x; f[1] = y*y; f[2] = z*z;
        f[3] = x*y; f[4] = x*z; f[5] = y*z;
        f[6] = x;   f[7] = y;   f[8] = z;
        f[9] = 1.0f; f[10] = 0.0f; f[11] = 0.0f;
    }
    __syncthreads();

    // ---- Phase 1: (128 x 12) x (12 x 16) GEMM per wave via V_WMMA_F32_16X16X4_F32
    const int lane = tid & 31;
    const int w    = tid >> 5;       // wave id 0..3
    const int col  = lane & 15;      // N (point) / M (center row) index
    const int hi   = lane >> 4;      // lane half -> K group / M+8 select
    const int pl   = w * 16 + col;   // point-local 0..63

    v2f b[3];
#pragma unroll
    for (int kk = 0; kk < 3; ++kk) {
        b[kk].x = sF[pl*KF + kk*4 + 2*hi + 0];
        b[kk].y = sF[pl*KF + kk*4 + 2*hi + 1];
    }

    v8f acc[8];
#pragma unroll
    for (int mt = 0; mt < 8; ++mt) {
        acc[mt] = (v8f){0.f,0.f,0.f,0.f,0.f,0.f,0.f,0.f};
#pragma unroll
        for (int kk = 0; kk < 3; ++kk) {
            v2f a;
            const int m = mt*16 + col;
            a.x = sQ[m*KF + kk*4 + 2*hi + 0];
            a.y = sQ[m*KF + kk*4 + 2*hi + 1];
            acc[mt] = __builtin_amdgcn_wmma_f32_16x16x4_f32(
                false, a, false, b[kk], (short)0, acc[mt], false, false);
        }
    }

    // exp transform + store tile: D vgpr e -> center mt*16 + 8*hi + e (consecutive)
#pragma unroll
    for (int mt = 0; mt < 8; ++mt) {
        float4 lo, hi4;
        lo.x  = 5.0f * __expf(-0.5f * acc[mt][0]);
        lo.y  = 5.0f * __expf(-0.5f * acc[mt][1]);
        lo.z  = 5.0f * __expf(-0.5f * acc[mt][2]);
        lo.w  = 5.0f * __expf(-0.5f * acc[mt][3]);
        hi4.x = 5.0f * __expf(-0.5f * acc[mt][4]);
        hi4.y = 5.0f * __expf(-0.5f * acc[mt][5]);
        hi4.z = 5.0f * __expf(-0.5f * acc[mt][6]);
        hi4.w = 5.0f * __expf(-0.5f * acc[mt][7]);
        const int base = pl * VSTR + mt*16 + 8*hi;
        *reinterpret_cast<float4*>(&sV[base + 0]) = lo;
        *reinterpret_cast<float4*>(&sV[base + 4]) = hi4;
    }
    __syncthreads();

    // ---- Phase 2: per-point selection + blend
    float pensum = 0.0f;
    if (tid < PPB) {
        const float* colv = &sV[tid * VSTR];
        const int p = pbase + tid;

        for (int i = 0; i < N_CEN; ++i)
            pensum += fmaxf(colv[i] - 0.01f, 0.0f);

        // tau = 16th largest value (max-extraction; ties measure-zero for random data)
        float tau = 3.0e38f;
        for (int t = 0; t < 16; ++t) {
            float m = -1.0f;
            for (int i = 0; i < N_CEN; ++i) {
                const float v = colv[i];
                m = (v < tau && v > m) ? v : m;
            }
            tau = m;
        }
        const float thr  = fmaxf(tau, 0.01f);
        const float dist = dists[p];

        float ws = 0.f, o0 = 0.f, o1 = 0.f, o2 = 0.f, o3 = 0.f;
        for (int i = 0; i < N_CEN; ++i) {
            const float v = colv[i];
            if (v >= thr) {
                const float4 rg =
                    reinterpret_cast<const float4*>(rgbs)[(size_t)i * N_PTS + p];
                const float alpha = 1.0f - __expf(-fmaxf(rg.w, 0.0f) * dist);
                ws += v;
                o0 += v * rg.x; o1 += v * rg.y; o2 += v * rg.z; o3 += v * alpha;
            }
        }
        const float inv = 1.0f / (ws + 1e-7f);
        float4 res = { o0*inv, o1*inv, o2*inv, o3*inv };
        reinterpret_cast<float4*>(out)[p] = res;

        sF[tid] = pensum;   // sF free after phase 1 -> reduce buffer
    }
    __syncthreads();

    // block reduce penalty over PPB entries, one atomic per block
    for (int s = PPB / 2; s > 0; s >>= 1) {
        if (tid < s) sF[tid] += sF[tid + s];
        __syncthreads();
    }
    if (tid == 0) atomicAdd(pen, sF[0] * (0.001f / (float)N_PTS));
}

// ---------------------------------------------------------------------------
extern "C" void kernel_launch(void* const* d_in, const int* in_sizes, int n_in,
                              void* d_out, int out_size, void* d_ws, size_t ws_size,
                              hipStream_t stream) {
    (void)in_sizes; (void)n_in; (void)out_size; (void)ws_size;
    const float* xyz   = (const float*)d_in[0];
    const float* cen   = (const float*)d_in[1];
    const float* rad   = (const float*)d_in[2];
    const float* rotv  = (const float*)d_in[3];
    const float* rgbs  = (const float*)d_in[4];
    const float* dists = (const float*)d_in[5];
    // d_in[6] = topk (int, ==16), fixed at compile time

    float* out = (float*)d_out;
    float* pen = out + (size_t)N_PTS * 4;   // tuple tail: scalar penalty
    float* Qw  = (float*)d_ws;              // 128*12 floats

    nerflet_prep<<<1, 128, 0, stream>>>(cen, rad, rotv, Qw, pen);
    nerflet_main<<<N_PTS / PPB, BLOCK, 0, stream>>>(xyz, Qw, rgbs, dists, out, pen);
}